// CoEvoGNN_1417339207928
// MI455X (gfx1250) — compile-verified
//
#include <hip/hip_runtime.h>
#include <math.h>

#define D 64
#define S 20
#define TILE 16
#define WAVES 8
#define NODES_PER_BLOCK (TILE * WAVES)

typedef float v2f __attribute__((ext_vector_type(2)));
typedef float v8f __attribute__((ext_vector_type(8)));

// One CoEvoGNN step: out = normalize(relu( sum_k [H_k | meanAgg_k(H_k)] @ W[k] ))
__global__ __launch_bounds__(256) void coevo_step(
    const float* __restrict__ prev1,   // H[t-1]
    const float* __restrict__ prev2,   // H[t-2] (ignored if kterms==1)
    const float* __restrict__ W,       // [K][2*D][D]
    const int*   __restrict__ idx1,    // neigh_idx[t-1]  [N][S]
    const int*   __restrict__ idx2,    // neigh_idx[t-2]
    float*       __restrict__ out,     // H[t]  [N][D]
    int N, int kterms)
{
    __shared__ float lds[WAVES][TILE * 2 * D];   // per-wave 16 x 128 cat tile (64 KB total)

    const int tid  = threadIdx.x;
    const int wave = tid >> 5;
    const int lane = tid & 31;
    const int half = lane >> 4;      // 0: K even-pair / rows 0-7 ; 1: K odd-pair / rows 8-15
    const int l16  = lane & 15;
    const int base = blockIdx.x * NODES_PER_BLOCK + wave * TILE;

    float* slice = &lds[wave][0];

    v8f acc0 = {}; v8f acc1 = {}; v8f acc2 = {}; v8f acc3 = {};

    for (int k = 0; k < kterms; ++k) {
        const float* Hk = (k == 0) ? prev1 : prev2;
        const int*   Ik = (k == 0) ? idx1  : idx2;
        const float* Wk = W + (size_t)k * (2 * D * D);

        __syncthreads();
        // ---- Phase 1: build cat = [self | mean-agg] for this wave's 16 nodes ----
        {
            const int d0 = lane, d1 = lane + 32;       // 2 dims per lane covers D=64
            for (int j = 0; j < TILE; ++j) {
                int node = base + j;
                if (node >= N) node = N - 1;           // clamp for safe reads (store is guarded)
                const float* hrow = Hk + (size_t)node * D;
                slice[j * 2 * D + d0] = hrow[d0];
                slice[j * 2 * D + d1] = hrow[d1];

                const int* ip = Ik + (size_t)node * S;
                float s0 = 0.f, s1 = 0.f;
                #pragma unroll 4
                for (int s = 0; s < S; ++s) {
                    const float* nr = Hk + (size_t)ip[s] * D;
                    s0 += nr[d0];
                    s1 += nr[d1];
                }
                slice[j * 2 * D + D + d0] = s0 * (1.0f / S);
                slice[j * 2 * D + D + d1] = s1 * (1.0f / S);
            }
        }
        __syncthreads();

        // ---- Phase 2: acc += cat(16x128) @ Wk(128x64) via V_WMMA_F32_16X16X4_F32 ----
        // A layout (16x4 f32): lanes 0-15 hold M=l16, K = kb+{0,1}; lanes 16-31 K = kb+{2,3}
        // B layout (4x16 f32): mirrored; C/D: vgpr i -> row half*8+i, col = ctile*16+l16
        for (int kc = 0; kc < (2 * D) / 4; ++kc) {
            const int kb = kc * 4 + half * 2;
            v2f a;
            a.x = slice[l16 * 2 * D + kb + 0];
            a.y = slice[l16 * 2 * D + kb + 1];

            const float* wb = Wk + (size_t)kb * D + l16;
            v2f b;
            b.x = wb[0];          b.y = wb[D];
            acc0 = __builtin_amdgcn_wmma_f32_16x16x4_f32(false, a, false, b, (short)0, acc0, false, false);
            b.x = wb[16];         b.y = wb[D + 16];
            acc1 = __builtin_amdgcn_wmma_f32_16x16x4_f32(false, a, false, b, (short)0, acc1, false, false);
            b.x = wb[32];         b.y = wb[D + 32];
            acc2 = __builtin_amdgcn_wmma_f32_16x16x4_f32(false, a, false, b, (short)0, acc2, false, false);
            b.x = wb[48];         b.y = wb[D + 48];
            acc3 = __builtin_amdgcn_wmma_f32_16x16x4_f32(false, a, false, b, (short)0, acc3, false, false);
        }
    }

    // ---- Phase 3: ReLU, row L2-norm (shfl butterfly within 16-lane halves), store ----
    float p[8];
    #pragma unroll
    for (int i = 0; i < 8; ++i) {
        float r0 = fmaxf(acc0[i], 0.f);
        float r1 = fmaxf(acc1[i], 0.f);
        float r2 = fmaxf(acc2[i], 0.f);
        float r3 = fmaxf(acc3[i], 0.f);
        acc0[i] = r0; acc1[i] = r1; acc2[i] = r2; acc3[i] = r3;
        p[i] = r0 * r0 + r1 * r1 + r2 * r2 + r3 * r3;
    }
    #pragma unroll
    for (int i = 0; i < 8; ++i) {
        float s = p[i];
        s += __shfl_xor(s, 1);
        s += __shfl_xor(s, 2);
        s += __shfl_xor(s, 4);
        s += __shfl_xor(s, 8);          // sum over the 16-lane half owning this row
        p[i] = 1.0f / fmaxf(sqrtf(s), 1e-12f);
    }
    #pragma unroll
    for (int i = 0; i < 8; ++i) {
        const int row = base + half * 8 + i;
        if (row < N) {
            float* orow = out + (size_t)row * D + l16;
            orow[0]  = acc0[i] * p[i];
            orow[16] = acc1[i] * p[i];
            orow[32] = acc2[i] * p[i];
            orow[48] = acc3[i] * p[i];
        }
    }
}

extern "C" void kernel_launch(void* const* d_in, const int* in_sizes, int n_in,
                              void* d_out, int out_size, void* d_ws, size_t ws_size,
                              hipStream_t stream) {
    (void)n_in; (void)d_ws; (void)ws_size; (void)out_size;

    const float* H0    = (const float*)d_in[0];   // [N, D]
    const float* W     = (const float*)d_in[1];   // [K, 2D, D]
    const int*   neigh = (const int*)d_in[2];     // [T, N, S]

    float* out = (float*)d_out;                   // H[1..5] stacked, [5, N, D]
    const int N = in_sizes[0] / D;
    const int T_TOTAL = 5;                        // t_train(4) + t_forecast(1)

    dim3 block(256);
    dim3 grid((N + NODES_PER_BLOCK - 1) / NODES_PER_BLOCK);

    for (int t = 1; t <= T_TOTAL; ++t) {
        const float* p1 = (t == 1) ? H0 : out + (size_t)(t - 2) * N * D;
        const float* p2 = (t == 2) ? H0
                        : (t >= 3) ? out + (size_t)(t - 3) * N * D
                                   : H0;                       // unused when t==1
        const int* i1 = neigh + (size_t)(t - 1) * N * S;
        const int* i2 = (t >= 2) ? neigh + (size_t)(t - 2) * N * S : i1;  // unused when t==1
        const int kterms = (t >= 2) ? 2 : 1;

        coevo_step<<<grid, block, 0, stream>>>(p1, p2, W, i1, i2,
                                               out + (size_t)(t - 1) * N * D,
                                               N, kterms);
    }
}